// Z2SeamGatedRNN_78700980731971
// MI455X (gfx1250) — compile-verified
//
#include <hip/hip_runtime.h>
#include <hip/hip_bf16.h>

typedef float v8f __attribute__((ext_vector_type(8)));
typedef float v2f __attribute__((ext_vector_type(2)));

#define T_  2048
#define D_  128

// D(16x16,f32) = A(16x4,f32) x B(4x16,f32) + C   -- fp32 WMMA, 8-arg form
__device__ __forceinline__ v8f wmma4(v2f a, v2f b, v8f c) {
  return __builtin_amdgcn_wmma_f32_16x16x4_f32(false, a, false, b, (short)0, c,
                                               false, false);
}

__device__ __forceinline__ float fast_tanh(float x) {
  float a = __builtin_fabsf(x);
  float e = __expf(-2.0f * a);
  float r = (1.0f - e) / (1.0f + e);
  return __builtin_copysignf(r, x);
}

// ---------------------------------------------------------------------------
// Kernel 1: x_proj[b,t,:] = x[b,t,:] @ W_in^T + (W_in_b + bias)
// One wave per (btile,t); output written in WMMA C-fragment order:
//   ws[((btile*T + t)*32 + lane)*32 + nt*8 + v]
// ---------------------------------------------------------------------------
__global__ void __launch_bounds__(256)
xproj_kernel(const float* __restrict__ x, const float* __restrict__ Wi,
             const float* __restrict__ Wib, const float* __restrict__ bias,
             float* __restrict__ xp) {
  const int lane = threadIdx.x & 31;
  const int wid  = blockIdx.x * 8 + (threadIdx.x >> 5);   // 0..32767
  const int btile = wid >> 11;
  const int t     = wid & 2047;
  const int lo = lane & 15, hf = lane >> 4;

  v8f acc[4];
#pragma unroll
  for (int nt = 0; nt < 4; ++nt) {
    float cst = Wib[nt * 16 + lo] + bias[nt * 16 + lo];
#pragma unroll
    for (int v = 0; v < 8; ++v) acc[nt][v] = cst;
  }

  const float* xrow = x + ((size_t)(btile * 16 + lo) * T_ + t) * D_ + hf * 2;
  const float* wrow = Wi + (size_t)lo * D_ + hf * 2;
#pragma unroll
  for (int ks = 0; ks < 32; ++ks) {
    v2f a = *(const v2f*)(xrow + ks * 4);
#pragma unroll
    for (int nt = 0; nt < 4; ++nt) {
      v2f b = *(const v2f*)(wrow + nt * 16 * D_ + ks * 4);
      acc[nt] = wmma4(a, b, acc[nt]);
    }
  }

  float* o = xp + ((size_t)wid * 32 + lane) * 32;
#pragma unroll
  for (int nt = 0; nt < 4; ++nt) {
#pragma unroll
    for (int q = 0; q < 2; ++q) {
      float4 s;
      s.x = acc[nt][q * 4 + 0]; s.y = acc[nt][q * 4 + 1];
      s.z = acc[nt][q * 4 + 2]; s.w = acc[nt][q * 4 + 3];
      *(float4*)(o + nt * 8 + q * 4) = s;
    }
  }
}

// ---------------------------------------------------------------------------
// Kernel 2: sequential scan. 16 blocks x 1 wave; each wave owns 16 batch rows.
// W_comm (block-diagonal) and seam B-fragments resident in VGPRs.
// h round-trips through 4KB LDS each step for the C->A layout change.
// ---------------------------------------------------------------------------
__global__ void __launch_bounds__(32)
scan_kernel(const float* __restrict__ Ap, const float* __restrict__ Am,
            const float* __restrict__ Wf, const float* __restrict__ Wo,
            const float* __restrict__ Wob, const float* __restrict__ xp,
            float* __restrict__ out) {
  __shared__ __align__(16) float hlds[16 * 64];
  const int lane = threadIdx.x;
  const int lo = lane & 15, hf = lane >> 4;
  const int btile = blockIdx.x;

  // W_comm^T fragments: block-diagonal -> ntile 0,1 need k-steps 0..7 (A_plus),
  // ntile 2,3 need k-steps 8..15 (A_minus).
  v2f bw[4][8];
#pragma unroll
  for (int nt = 0; nt < 4; ++nt) {
    const float* src = (nt < 2) ? Ap : Am;
    const int j = nt * 16 + lo;
#pragma unroll
    for (int s = 0; s < 8; ++s) {
      const int ks = (nt < 2) ? s : (s + 8);
      bw[nt][s] = *(const v2f*)(src + j * 64 + ks * 4 + hf * 2);
    }
  }
  // seam^T fragments: seam = S @ W_flip -> sign-flip rows j>=32 (ntile 2,3).
  v2f bs[4][16];
#pragma unroll
  for (int nt = 0; nt < 4; ++nt) {
    const int j = nt * 16 + lo;
    const float sg = (nt < 2) ? 1.0f : -1.0f;
#pragma unroll
    for (int ks = 0; ks < 16; ++ks) {
      v2f w = *(const v2f*)(Wf + j * 64 + ks * 4 + hf * 2);
      v2f r; r.x = w.x * sg; r.y = w.y * sg;
      bs[nt][ks] = r;
    }
  }

  // h0 = 0
  for (int i = lane; i < 1024; i += 32) hlds[i] = 0.0f;
  __syncthreads();

  const float* xpl = xp + (size_t)btile * T_ * 1024 + lane * 32;
  float4 xq[8];
#pragma unroll
  for (int q = 0; q < 8; ++q) xq[q] = *(const float4*)(xpl + q * 4);

#pragma unroll 1
  for (int t = 0; t < T_; ++t) {
    // Seed accumulators with prefetched x_proj (bias/W_in_b already folded in).
    v8f uc[4], u2[4], u3[4];
#pragma unroll
    for (int nt = 0; nt < 4; ++nt) {
      float4 q0 = xq[nt * 2], q1 = xq[nt * 2 + 1];
      uc[nt][0] = q0.x; uc[nt][1] = q0.y; uc[nt][2] = q0.z; uc[nt][3] = q0.w;
      uc[nt][4] = q1.x; uc[nt][5] = q1.y; uc[nt][6] = q1.z; uc[nt][7] = q1.w;
#pragma unroll
      for (int v = 0; v < 8; ++v) { u2[nt][v] = 0.0f; u3[nt][v] = 0.0f; }
    }
    // Prefetch next step's x_proj while WMMAs run.
    if (t + 1 < T_) {
      const float* nx = xpl + (size_t)(t + 1) * 1024;
#pragma unroll
      for (int q = 0; q < 8; ++q) xq[q] = *(const float4*)(nx + q * 4);
      __builtin_prefetch(xpl + (size_t)(t + 5) * 1024, 0, 1);
    }

    // h_t in A-matrix layout: lane holds row=lo, k = ks*4 + hf*2 + {0,1}.
    v2f ha[16];
#pragma unroll
    for (int ks = 0; ks < 16; ++ks)
      ha[ks] = *(const v2f*)&hlds[lo * 64 + ks * 4 + hf * 2];

    // Parity gate: alpha = ||h[k>=32]||^2 / (||h||^2 + eps), per row.
    float pa = 0.0f, pm = 0.0f;
#pragma unroll
    for (int ks = 0; ks < 16; ++ks) {
      float d = ha[ks].x * ha[ks].x + ha[ks].y * ha[ks].y;
      pa += d;
      if (ks >= 8) pm += d;   // k>=32  <=>  ks>=8
    }
    pa += __shfl_xor(pa, 16, 32);   // partner lane holds the other 32 k's
    pm += __shfl_xor(pm, 16, 32);
    const float alpha = pm / (pa + 1e-8f);
    const float g =
        1.0f / (1.0f + __expf(-(alpha - 0.72134752044448170f) * 20.0f));
    v2f hg[16];
#pragma unroll
    for (int ks = 0; ks < 16; ++ks) {
      hg[ks].x = ha[ks].x * g; hg[ks].y = ha[ks].y * g;
    }

    // u = h@W_comm^T (block-diag, depth 8) + g*h@seam^T (two depth-8 chains)
#pragma unroll
    for (int nt = 0; nt < 4; ++nt) {
#pragma unroll
      for (int s = 0; s < 8; ++s)
        uc[nt] = wmma4(ha[(nt < 2) ? s : (s + 8)], bw[nt][s], uc[nt]);
#pragma unroll
      for (int s = 0; s < 8; ++s)
        u2[nt] = wmma4(hg[s], bs[nt][s], u2[nt]);
#pragma unroll
      for (int s = 8; s < 16; ++s)
        u3[nt] = wmma4(hg[s], bs[nt][s], u3[nt]);
    }

    __syncthreads();
#pragma unroll
    for (int nt = 0; nt < 4; ++nt)
#pragma unroll
      for (int v = 0; v < 8; ++v)
        hlds[(v + 8 * hf) * 64 + nt * 16 + lo] =
            fast_tanh(uc[nt][v] + u2[nt][v] + u3[nt][v]);
    __syncthreads();
  }

  // hT -> d_out[32768 + b*64 + h]
#pragma unroll 1
  for (int i = lane; i < 1024; i += 32)
    out[32768 + btile * 1024 + i] = hlds[i];

  // predictions = hT @ W_out^T + W_out_b  -> d_out[0 .. 32768)
  v2f ph[16];
#pragma unroll
  for (int ks = 0; ks < 16; ++ks)
    ph[ks] = *(const v2f*)&hlds[lo * 64 + ks * 4 + hf * 2];
#pragma unroll
  for (int n2 = 0; n2 < 8; ++n2) {
    v8f p;
    const float c0 = Wob[n2 * 16 + lo];
#pragma unroll
    for (int v = 0; v < 8; ++v) p[v] = c0;
#pragma unroll
    for (int ks = 0; ks < 16; ++ks) {
      v2f b = *(const v2f*)(Wo + (size_t)(n2 * 16 + lo) * 64 + ks * 4 + hf * 2);
      p = wmma4(ph[ks], b, p);
    }
#pragma unroll
    for (int v = 0; v < 8; ++v)
      out[(size_t)(btile * 16 + v + 8 * hf) * 128 + n2 * 16 + lo] = p[v];
  }
}

extern "C" void kernel_launch(void* const* d_in, const int* in_sizes, int n_in,
                              void* d_out, int out_size, void* d_ws, size_t ws_size,
                              hipStream_t stream) {
  const float* x    = (const float*)d_in[0];
  const float* Ap   = (const float*)d_in[1];
  const float* Am   = (const float*)d_in[2];
  const float* Wf   = (const float*)d_in[3];
  const float* Wiw  = (const float*)d_in[4];
  const float* Wib  = (const float*)d_in[5];
  const float* Wow  = (const float*)d_in[6];
  const float* Wob  = (const float*)d_in[7];
  const float* bias = (const float*)d_in[8];
  float* out = (float*)d_out;
  float* xp  = (float*)d_ws;   // 16*2048*16*64 f32 = 128 MiB fragment-ordered x_proj

  xproj_kernel<<<4096, 256, 0, stream>>>(x, Wiw, Wib, bias, xp);
  scan_kernel<<<16, 32, 0, stream>>>(Ap, Am, Wf, Wow, Wob, xp, out);
}